// EvolveGCNH_2199023255948
// MI455X (gfx1250) — compile-verified
//
#include <hip/hip_runtime.h>
#include <hip/hip_bf16.h>
#include <math.h>

typedef __attribute__((ext_vector_type(2))) float v2f;
typedef __attribute__((ext_vector_type(8))) float v8f;

#define D_C       64
#define TOPK_C    32
#define NPAR      4160           // 64*64 + 64
#define GROWS     (3 * NPAR)     // 12480
#define GIN       (TOPK_C * D_C) // 2048

// ---------------------------------------------------------------- small utils
__device__ __forceinline__ float wave_reduce_add(float v) {
    #pragma unroll
    for (int off = 16; off; off >>= 1) v += __shfl_down(v, off, 32);
    return v;
}

// ||p|| + eps  (single wave, 64 elements)
__global__ void pnorm_kernel(const float* __restrict__ p, float* __restrict__ pn) {
    int lane = threadIdx.x;
    float a = p[lane], b = p[lane + 32];
    float s = wave_reduce_add(a * a + b * b);
    if (lane == 0) pn[0] = sqrtf(s) + 1e-8f;
}

// y[i] = dot(h[i], p) / pnorm   — one wave per node, float2 (b64) coalesced reads
__global__ __launch_bounds__(256) void score_kernel(
    const float* __restrict__ h, const float* __restrict__ p,
    const float* __restrict__ pn, float* __restrict__ y, int n) {
    int node = blockIdx.x * (blockDim.x >> 5) + (threadIdx.x >> 5);
    if (node >= n) return;
    int lane = threadIdx.x & 31;
    const float2* hr = (const float2*)(h + (size_t)node * D_C);
    const float2* pp = (const float2*)p;
    float2 a = hr[lane], b = pp[lane];
    float acc = wave_reduce_add(a.x * b.x + a.y * b.y);
    if (lane == 0) y[node] = acc / pn[0];
}

// top-32 by repeated block-wide argmax on packed sortable keys (destroys y).
__global__ __launch_bounds__(1024) void topk32_kernel(
    float* __restrict__ y, int n, int* __restrict__ idx_out, float* __restrict__ w_out) {
    __shared__ unsigned long long red[1024];
    int t = threadIdx.x;
    for (int iter = 0; iter < TOPK_C; ++iter) {
        unsigned long long best = 0ull;
        for (int i = t; i < n; i += blockDim.x) {
            unsigned u = __float_as_uint(y[i]);
            u = (u & 0x80000000u) ? ~u : (u | 0x80000000u);   // monotonic key
            unsigned long long key =
                ((unsigned long long)u << 32) | (unsigned)(n - 1 - i); // ties -> low idx
            if (key > best) best = key;
        }
        red[t] = best;
        __syncthreads();
        for (int s = 512; s > 0; s >>= 1) {
            if (t < s && red[t + s] > red[t]) red[t] = red[t + s];
            __syncthreads();
        }
        if (t == 0) {
            int i = n - 1 - (int)(red[0] & 0xffffffffu);
            idx_out[iter] = i;
            w_out[iter]   = tanhf(y[i]);
            y[i] = -3.4e38f;
        }
        __syncthreads();
    }
}

// z_flat[d*32+k] = h[idx[k]][d] * w[k]   (Z.T flatten)
__global__ void build_z_kernel(const float* __restrict__ h, const int* __restrict__ idx,
                               const float* __restrict__ w, float* __restrict__ z) {
    int t = blockIdx.x * blockDim.x + threadIdx.x;
    if (t >= GIN) return;
    int k = t & 31, d = t >> 5;
    z[t] = h[(size_t)idx[k] * D_C + d] * w[k];
}

// h0 = concat(w.flat, b)
__global__ void build_h0_kernel(const float* __restrict__ w, const float* __restrict__ b,
                                float* __restrict__ h0) {
    int t = blockIdx.x * blockDim.x + threadIdx.x;
    if (t < D_C * D_C) h0[t] = w[t];
    else if (t < NPAR) h0[t] = b[t - D_C * D_C];
}

// out[r] = dot(M[r,:], v) + bias[r] — one wave per row, float4 (b128) loads.
// This is the dominant HBM stream (~640 MB of GRU weights per call).
__global__ __launch_bounds__(256) void gemv_kernel(
    const float* __restrict__ M, const float* __restrict__ v,
    const float* __restrict__ bias, float* __restrict__ out, int rows, int cols) {
    int row = blockIdx.x * (blockDim.x >> 5) + (threadIdx.x >> 5);
    if (row >= rows) return;
    int lane = threadIdx.x & 31;
    const float4* mr = (const float4*)(M + (size_t)row * cols);
    const float4* vv = (const float4*)v;
    int n4 = cols >> 2;
    float acc = 0.f;
    for (int c = lane; c < n4; c += 32) {
        float4 m = mr[c], x = vv[c];
        acc += m.x * x.x + m.y * x.y + m.z * x.z + m.w * x.w;
    }
    acc = wave_reduce_add(acc);
    if (lane == 0) out[row] = acc + bias[row];
}

// GRU gate fusion -> evolved params nw[0..4159]
__global__ void gru_combine_kernel(const float* __restrict__ gi, const float* __restrict__ gh,
                                   const float* __restrict__ h0, float* __restrict__ nw) {
    int i = blockIdx.x * blockDim.x + threadIdx.x;
    if (i >= NPAR) return;
    float r = 1.f / (1.f + expf(-(gi[i] + gh[i])));
    float z = 1.f / (1.f + expf(-(gi[NPAR + i] + gh[NPAR + i])));
    float nn = tanhf(gi[2 * NPAR + i] + r * gh[2 * NPAR + i]);
    nw[i] = (1.f - z) * nn + z * h0[i];
}

// xw = h (Nx64) @ weight^T (64x64) using V_WMMA_F32_16X16X4_F32.
// Each wave: one 16-row tile, full N=64 via four 16x16 f32 accumulators.
// Weight (16KB) staged in LDS; compiler hoists all B fragments into VGPRs.
__global__ __launch_bounds__(256) void gcn_xw_wmma_kernel(
    const float* __restrict__ h, const float* __restrict__ weight,
    float* __restrict__ xw, int n) {
    __shared__ float lw[D_C * D_C];
    for (int i = threadIdx.x; i < D_C * D_C; i += blockDim.x) lw[i] = weight[i];
    __syncthreads();

    const int lane = threadIdx.x & 31;
    const int wv   = threadIdx.x >> 5;
    const int half = lane >> 4;        // K-pair selector (0: K0/1, 1: K2/3)
    const int m16  = lane & 15;        // row within A tile / col within B,C tiles
    const int tiles = n >> 4;          // n divisible by 16 (100000 = 6250*16)
    const int wavesPerGrid = gridDim.x * (blockDim.x >> 5);

    for (int tile = blockIdx.x * (blockDim.x >> 5) + wv; tile < tiles; tile += wavesPerGrid) {
        const int m0 = tile << 4;
        const float* hrow = h + (size_t)(m0 + m16) * D_C;
        v8f c0 = {}, c1 = {}, c2 = {}, c3 = {};
        #pragma unroll
        for (int k = 0; k < D_C; k += 4) {
            const int kk = k + 2 * half;
            v2f a; a[0] = hrow[kk]; a[1] = hrow[kk + 1];            // A 16x4 fragment
            v2f b0, b1, b2, b3;                                     // B 4x16 fragments
            b0[0] = lw[(m16     ) * D_C + kk]; b0[1] = lw[(m16     ) * D_C + kk + 1];
            b1[0] = lw[(m16 + 16) * D_C + kk]; b1[1] = lw[(m16 + 16) * D_C + kk + 1];
            b2[0] = lw[(m16 + 32) * D_C + kk]; b2[1] = lw[(m16 + 32) * D_C + kk + 1];
            b3[0] = lw[(m16 + 48) * D_C + kk]; b3[1] = lw[(m16 + 48) * D_C + kk + 1];
            c0 = __builtin_amdgcn_wmma_f32_16x16x4_f32(false, a, false, b0, (short)0, c0, false, false);
            c1 = __builtin_amdgcn_wmma_f32_16x16x4_f32(false, a, false, b1, (short)0, c1, false, false);
            c2 = __builtin_amdgcn_wmma_f32_16x16x4_f32(false, a, false, b2, (short)0, c2, false, false);
            c3 = __builtin_amdgcn_wmma_f32_16x16x4_f32(false, a, false, b3, (short)0, c3, false, false);
        }
        // C/D layout: VGPR r -> row (r + 8*half), col = m16 (+16 per N tile)
        #pragma unroll
        for (int r = 0; r < 8; ++r) {
            float* orow = xw + (size_t)(m0 + r + 8 * half) * D_C + m16;
            orow[0]  = c0[r];
            orow[16] = c1[r];
            orow[32] = c2[r];
            orow[48] = c3[r];
        }
    }
}

// ----------------------------------------------------------- graph aggregation
__global__ void deg_init_kernel(float* __restrict__ deg, int n) {
    int i = blockIdx.x * blockDim.x + threadIdx.x;
    if (i < n) deg[i] = 1.f;                     // self loop
}
__global__ void deg_accum_kernel(const long long* __restrict__ ei, float* __restrict__ deg, int ne) {
    int e = blockIdx.x * blockDim.x + threadIdx.x;
    if (e < ne) atomicAdd(&deg[(int)ei[(size_t)ne + e]], 1.f);
}
__global__ void dinv_kernel(float* __restrict__ deg, int n) {
    int i = blockIdx.x * blockDim.x + threadIdx.x;
    if (i < n) deg[i] = rsqrtf(deg[i]);          // deg >= 1 always
}
// out[v] = xw[v]*dinv[v]^2 + bias  (self loop + bias; fully overwrites out). float4.
__global__ void self_init_kernel(const float* __restrict__ xw, const float* __restrict__ dinv,
                                 const float* __restrict__ bias, float* __restrict__ out, int n) {
    int t = blockIdx.x * blockDim.x + threadIdx.x;
    if (t >= n * (D_C / 4)) return;
    int v = t >> 4, q = (t & 15) << 2;
    float di2 = dinv[v]; di2 *= di2;
    float4 m = *(const float4*)(xw + (size_t)v * D_C + q);
    float4 bb = *(const float4*)(bias + q);
    float4 o;
    o.x = m.x * di2 + bb.x; o.y = m.y * di2 + bb.y;
    o.z = m.z * di2 + bb.z; o.w = m.w * di2 + bb.w;
    *(float4*)(out + (size_t)v * D_C + q) = o;
}
// scatter-add messages; 16 threads/edge, float4 gathers (b128), 4 atomics each.
// xw + out (~51 MB) are L2-resident, so atomics stay on-chip.
__global__ __launch_bounds__(256) void edge_aggr_kernel(
    const long long* __restrict__ ei, const float* __restrict__ xw,
    const float* __restrict__ dinv, float* __restrict__ out, int ne) {
    int total = ne * 16;
    int stride = gridDim.x * blockDim.x;
    for (int t = blockIdx.x * blockDim.x + threadIdx.x; t < total; t += stride) {
        int e = t >> 4, q = (t & 15) << 2;
        int s = (int)ei[e];
        int d = (int)ei[(size_t)ne + e];
        float nrm = dinv[s] * dinv[d];
        float4 m = *(const float4*)(xw + (size_t)s * D_C + q);
        float* o = out + (size_t)d * D_C + q;
        atomicAdd(o + 0, m.x * nrm);
        atomicAdd(o + 1, m.y * nrm);
        atomicAdd(o + 2, m.z * nrm);
        atomicAdd(o + 3, m.w * nrm);
    }
}
__global__ void relu_kernel(float* __restrict__ x, int n4) {
    int i = blockIdx.x * blockDim.x + threadIdx.x;
    if (i >= n4) return;
    float4* p = (float4*)x + i;
    float4 v = *p;
    v.x = fmaxf(v.x, 0.f); v.y = fmaxf(v.y, 0.f);
    v.z = fmaxf(v.z, 0.f); v.w = fmaxf(v.w, 0.f);
    *p = v;
}

// ---------------------------------------------------------------- host driver
struct Scratch {
    float* y; float* dinv; float* pn; float* zf; float* h0; float* gi; float* gh;
    float* nw; float* xw; float* h1; int* idx; float* wtanh;
};

static void run_layer(const float* h_in, float* h_out,
                      const float* p, const float* w, const float* b,
                      const float* wih, const float* whh,
                      const float* bih, const float* bhh,
                      const long long* ei, int n, int ne, bool relu,
                      const Scratch& S, hipStream_t stream) {
    pnorm_kernel<<<1, 32, 0, stream>>>(p, S.pn);
    score_kernel<<<(n + 7) / 8, 256, 0, stream>>>(h_in, p, S.pn, S.y, n);
    topk32_kernel<<<1, 1024, 0, stream>>>(S.y, n, S.idx, S.wtanh);
    build_z_kernel<<<(GIN + 255) / 256, 256, 0, stream>>>(h_in, S.idx, S.wtanh, S.zf);
    build_h0_kernel<<<(NPAR + 255) / 256, 256, 0, stream>>>(w, b, S.h0);
    gemv_kernel<<<(GROWS + 7) / 8, 256, 0, stream>>>(wih, S.zf, bih, S.gi, GROWS, GIN);
    gemv_kernel<<<(GROWS + 7) / 8, 256, 0, stream>>>(whh, S.h0, bhh, S.gh, GROWS, NPAR);
    gru_combine_kernel<<<(NPAR + 255) / 256, 256, 0, stream>>>(S.gi, S.gh, S.h0, S.nw);
    // evolved weight = S.nw[0:4096] (64x64 row-major), bias = S.nw + 4096
    gcn_xw_wmma_kernel<<<(n / 16 + 7) / 8, 256, 0, stream>>>(h_in, S.nw, S.xw, n);
    self_init_kernel<<<(n * 16 + 255) / 256, 256, 0, stream>>>(S.xw, S.dinv, S.nw + D_C * D_C, h_out, n);
    edge_aggr_kernel<<<4096, 256, 0, stream>>>(ei, S.xw, S.dinv, h_out, ne);
    if (relu) relu_kernel<<<(n * 16 + 255) / 256, 256, 0, stream>>>(h_out, n * 16);
}

extern "C" void kernel_launch(void* const* d_in, const int* in_sizes, int n_in,
                              void* d_out, int out_size, void* d_ws, size_t ws_size,
                              hipStream_t stream) {
    const float*     x    = (const float*)d_in[0];
    const long long* ei   = (const long long*)d_in[1];
    const float*     p0   = (const float*)d_in[2];
    const float*     p1   = (const float*)d_in[3];
    const float*     w0   = (const float*)d_in[4];
    const float*     b0   = (const float*)d_in[5];
    const float*     w1   = (const float*)d_in[6];
    const float*     b1   = (const float*)d_in[7];
    const float*     wih0 = (const float*)d_in[8];
    const float*     whh0 = (const float*)d_in[9];
    const float*     bih0 = (const float*)d_in[10];
    const float*     bhh0 = (const float*)d_in[11];
    const float*     wih1 = (const float*)d_in[12];
    const float*     whh1 = (const float*)d_in[13];
    const float*     bih1 = (const float*)d_in[14];
    const float*     bhh1 = (const float*)d_in[15];

    const int n  = in_sizes[0] / D_C;   // 100000
    const int ne = in_sizes[1] / 2;     // 1600000

    float* f = (float*)d_ws;
    Scratch S;
    size_t off = 0;
    S.y    = f + off; off += (size_t)n;            // scores
    S.dinv = f + off; off += (size_t)n;            // deg -> dinv
    S.pn   = f + off; off += 32;                   // |p|
    S.zf   = f + off; off += GIN;                  // 2048
    S.h0   = f + off; off += NPAR;                 // 4160
    S.gi   = f + off; off += GROWS;                // 12480
    S.gh   = f + off; off += GROWS;                // 12480
    S.nw   = f + off; off += NPAR;                 // evolved params
    off = (off + 63) & ~(size_t)63;
    S.xw   = f + off; off += (size_t)n * D_C;      // 6.4M floats
    S.h1   = f + off; off += (size_t)n * D_C;      // 6.4M floats
    S.idx  = (int*)(f + off); off += 32;
    S.wtanh = f + off; off += 32;                  // ~52.1 MB total

    // degree / normalization (edge structure shared by both layers)
    deg_init_kernel<<<(n + 255) / 256, 256, 0, stream>>>(S.dinv, n);
    deg_accum_kernel<<<(ne + 255) / 256, 256, 0, stream>>>(ei, S.dinv, ne);
    dinv_kernel<<<(n + 255) / 256, 256, 0, stream>>>(S.dinv, n);

    // layer 0: x -> h1 (relu)
    run_layer(x, S.h1, p0, w0, b0, wih0, whh0, bih0, bhh0, ei, n, ne, true, S, stream);
    // layer 1: h1 -> d_out (no relu)
    run_layer(S.h1, (float*)d_out, p1, w1, b1, wih1, whh1, bih1, bhh1, ei, n, ne, false, S, stream);
}